// neural_network_16106127360589
// MI455X (gfx1250) — compile-verified
//
#include <hip/hip_runtime.h>

typedef __attribute__((ext_vector_type(2))) float v2f;
typedef __attribute__((ext_vector_type(8))) float v8f;

#define T_LEN 2048
#define HID 5
#define NS 2   // independent batch-row streams per wave (ILP for the serial chain)

// ---------- fast activations (native v_tanh_f32 on gfx1250) ----------
__device__ __forceinline__ float fast_sigmoid(float y) {
#if __has_builtin(__builtin_amdgcn_tanhf)
  return fmaf(0.5f, __builtin_amdgcn_tanhf(0.5f * y), 0.5f);
#elif __has_builtin(__builtin_amdgcn_exp2f)
  float e = __builtin_amdgcn_exp2f(-1.44269504088896341f * y);  // e^-y
  return __builtin_amdgcn_rcpf(1.0f + e);
#else
  float e = __expf(-y);
  return __builtin_amdgcn_rcpf(1.0f + e);
#endif
}

__device__ __forceinline__ float fast_tanh(float y) {
#if __has_builtin(__builtin_amdgcn_tanhf)
  return __builtin_amdgcn_tanhf(y);
#else
  return fmaf(2.0f, fast_sigmoid(2.0f * y), -1.0f);
#endif
}

// cross-half swap: lane <-> lane^16 (ds_swizzle group-of-32, xor_mask=0x10)
__device__ __forceinline__ float swap16(float v) {
  return __int_as_float(__builtin_amdgcn_ds_swizzle(__float_as_int(v), 0x401F));
}

// Padded weight matrix W' [32 x 8]:
//   row = 8*gate + j (gate: 0=i,1=f,2=g,3=o; j=0..4 real, 5..7 zero-pad)
//   col k: 0..4 -> W_hh[5g+j, k], 5 -> W_ih[5g+j], 6 -> b_ih+b_hh, 7 -> 0
__device__ __forceinline__ float wprime(int row, int k,
                                        const float* __restrict__ Whh,
                                        const float* __restrict__ Wih,
                                        const float* __restrict__ bih,
                                        const float* __restrict__ bhh) {
  int g = row >> 3, j = row & 7;
  if (j >= HID) return 0.0f;
  int r5 = 5 * g + j;
  if (k < HID)  return Whh[r5 * HID + k];
  if (k == 5)   return Wih[r5];
  if (k == 6)   return bih[r5] + bhh[r5];
  return 0.0f;
}

__global__ __launch_bounds__(128)
__attribute__((amdgpu_waves_per_eu(1)))
void lstm_wmma_f32_kernel(
    const float* __restrict__ x,   // [B, T, 1]
    const float* __restrict__ Wih, // [20, 1]
    const float* __restrict__ Whh, // [20, 5]
    const float* __restrict__ bih, // [20]
    const float* __restrict__ bhh, // [20]
    const float* __restrict__ Wfc, // [1, 5]
    const float* __restrict__ bfc, // [1]
    float* __restrict__ out)       // [B, 1]
{
  const int lane = threadIdx.x & 31;
  const int wave = threadIdx.x >> 5;
  const bool lo  = lane < 16;
  const int  mr  = lane & 15;                           // batch column within tile
  const int  m0  = (blockIdx.x * 4 + wave) * (16 * NS); // 4 waves/block, 32 rows/wave

  // ---- constant A tiles (weights), shared by both streams.
  const int kb = lo ? 0 : 2;
  v2f A00 = { wprime(mr,      kb,     Whh, Wih, bih, bhh),
              wprime(mr,      kb + 1, Whh, Wih, bih, bhh) };   // tile0 (i|f), K 0..3
  v2f A01 = { wprime(mr,      kb + 4, Whh, Wih, bih, bhh),
              wprime(mr,      kb + 5, Whh, Wih, bih, bhh) };   // tile0, K 4..7
  v2f A10 = { wprime(mr + 16, kb,     Whh, Wih, bih, bhh),
              wprime(mr + 16, kb + 1, Whh, Wih, bih, bhh) };   // tile1 (g|o), K 0..3
  v2f A11 = { wprime(mr + 16, kb + 4, Whh, Wih, bih, bhh),
              wprime(mr + 16, kb + 5, Whh, Wih, bih, bhh) };   // tile1, K 4..7

  // parametric activation  a(x) = Mul * sigmoid(Arg*x) + Add :
  //  self path : low half tanh (g-gates), high half sigmoid (o-gates)
  //  cross path: low half sigmoid (o-gates), high half tanh (g-gates)
  const float sArg = lo ? 2.0f : 1.0f, sMul = lo ? 2.0f : 1.0f, sAdd = lo ? -1.0f : 0.0f;
  const float cArg = lo ? 1.0f : 2.0f, cMul = lo ? 1.0f : 2.0f, cAdd = lo ?  0.0f : -1.0f;

  // State carried directly in B-operand layout (no selects on the feedback path):
  //   bh[0] = lo ? h0 : h2   (== B0.x)     bc[0] = lo ? c0 : c2
  //   bh[1] = lo ? h1 : h3   (== B0.y)     bc[1] = lo ? c1 : c3
  //   bh[2] = h4 on low half (high half: don't-care, masked at B build)
  float bh[NS][3], bc[NS][3];
#pragma unroll
  for (int s = 0; s < NS; ++s)
#pragma unroll
    for (int k = 0; k < 3; ++k) { bh[s][k] = 0.0f; bc[s][k] = 0.0f; }

  const float4* __restrict__ xrow[NS];
#pragma unroll
  for (int s = 0; s < NS; ++s)
    xrow[s] = (const float4*)(x + (size_t)(m0 + s * 16 + mr) * (size_t)T_LEN);

  for (int t4 = 0; t4 < T_LEN / 4; ++t4) {
    float xs[NS][4];
#pragma unroll
    for (int s = 0; s < NS; ++s) {
      float4 xq = xrow[s][t4];
      __builtin_prefetch(&xrow[s][t4 + 8], 0, 3);  // global_prefetch_b8, 32 steps ahead
      xs[s][0] = xq.x; xs[s][1] = xq.y; xs[s][2] = xq.z; xs[s][3] = xq.w;
    }

#pragma unroll
    for (int u = 0; u < 4; ++u) {
      v8f acc0[NS], acc1[NS];

      // --- issue both streams' WMMAs together (independent chains) ---
#pragma unroll
      for (int s = 0; s < NS; ++s) {
        const float xt = xs[s][u];
        v2f B0 = { bh[s][0], bh[s][1] };                    // pure register carry!
        v2f B1 = { lo ? bh[s][2] : 1.0f, lo ? xt : 0.0f };  // [h4|1, x|0]
        v8f z = {};
        acc0[s] = __builtin_amdgcn_wmma_f32_16x16x4_f32(false, A00, false, B0,
                                                        (short)0, z, false, false);
        acc0[s] = __builtin_amdgcn_wmma_f32_16x16x4_f32(false, A01, false, B1,
                                                        (short)0, acc0[s], false, false);
        acc1[s] = __builtin_amdgcn_wmma_f32_16x16x4_f32(false, A10, false, B0,
                                                        (short)0, z, false, false);
        acc1[s] = __builtin_amdgcn_wmma_f32_16x16x4_f32(false, A11, false, B1,
                                                        (short)0, acc1[s], false, false);
      }

      // acc0[s][r]: i_r (low lanes) | f_r (high lanes)
      // acc1[s][r]: g_r (low lanes) | o_r (high lanes)
#pragma unroll
      for (int s = 0; s < NS; ++s) {
        // fused pairs: low half processes r=p, high half processes r=p+2
#pragma unroll
        for (int p = 0; p < 2; ++p) {
          // Extract to scalars and pin them in VGPRs with a zero-cost asm
          // barrier: blocks the select(extract,extract) -> dynamic-extract
          // fold that produced a v_cmp/v_cndmask ladder in round-4 asm.
          float aP = acc0[s][p];
          float aQ = acc0[s][p + 2];
          float gP = acc1[s][p];
          float gQ = acc1[s][p + 2];
          asm("" : "+v"(aP), "+v"(aQ), "+v"(gP), "+v"(gQ));

          float a_self  = lo ? aP : aQ;
          float a_cross = swap16(lo ? aQ : aP);   // partner half's i/f gate
          float g_self  = lo ? gP : gQ;
          float g_cross = swap16(lo ? gQ : gP);   // partner half's g/o gate

          float s_self  = fast_sigmoid(a_self);
          float s_cross = fast_sigmoid(a_cross);
          float t_self  = fmaf(sMul, fast_sigmoid(sArg * g_self),  sAdd); // tanh|sig
          float t_cross = fmaf(cMul, fast_sigmoid(cArg * g_cross), cAdd); // sig|tanh
          float iv = lo ? s_self  : s_cross;
          float fv = lo ? s_cross : s_self;
          float gv = lo ? t_self  : t_cross;
          float ov = lo ? t_cross : t_self;
          bc[s][p] = fmaf(fv, bc[s][p], iv * gv);
          bh[s][p] = ov * fast_tanh(bc[s][p]);   // == next step's B0 component
        }
        // r = 4 (low half only; high half result is bounded garbage, masked later)
        {
          float i4 = fast_sigmoid(acc0[s][4]);
          float f4 = fast_sigmoid(swap16(acc0[s][4]));
          float g4 = fast_tanh(acc1[s][4]);
          float o4 = fast_sigmoid(swap16(acc1[s][4]));
          bc[s][2] = fmaf(f4, bc[s][2], i4 * g4);
          bh[s][2] = o4 * fast_tanh(bc[s][2]);
        }
      }
    }
  }

  // Gather high-half h2,h3 before the divergent store (EXEC still all ones here).
  float h2g[NS], h3g[NS];
#pragma unroll
  for (int s = 0; s < NS; ++s) { h2g[s] = swap16(bh[s][0]); h3g[s] = swap16(bh[s][1]); }

  // final FC: out = relu(h_T) @ W_fc^T + b_fc  (low half owns unique rows)
  if (lo) {
#pragma unroll
    for (int s = 0; s < NS; ++s) {
      float acc = bfc[0];
      acc = fmaf(fmaxf(bh[s][0], 0.0f), Wfc[0], acc);  // h0
      acc = fmaf(fmaxf(bh[s][1], 0.0f), Wfc[1], acc);  // h1
      acc = fmaf(fmaxf(h2g[s],   0.0f), Wfc[2], acc);  // h2 (from high half)
      acc = fmaf(fmaxf(h3g[s],   0.0f), Wfc[3], acc);  // h3 (from high half)
      acc = fmaf(fmaxf(bh[s][2], 0.0f), Wfc[4], acc);  // h4
      out[m0 + s * 16 + mr] = acc;
    }
  }
}

extern "C" void kernel_launch(void* const* d_in, const int* in_sizes, int n_in,
                              void* d_out, int out_size, void* d_ws, size_t ws_size,
                              hipStream_t stream) {
  (void)n_in; (void)out_size; (void)d_ws; (void)ws_size;
  const float* x   = (const float*)d_in[0];
  const float* Wih = (const float*)d_in[1];
  const float* Whh = (const float*)d_in[2];
  const float* bih = (const float*)d_in[3];
  const float* bhh = (const float*)d_in[4];
  const float* Wfc = (const float*)d_in[5];
  const float* bfc = (const float*)d_in[6];
  float* out = (float*)d_out;

  const int B = in_sizes[0] / T_LEN;              // 8192
  const int rowsPerBlock = 4 * 16 * NS;           // 4 waves x 32 batch rows
  const int grid = (B + rowsPerBlock - 1) / rowsPerBlock;   // 64 blocks

  lstm_wmma_f32_kernel<<<grid, 128, 0, stream>>>(x, Wih, Whh, bih, bhh, Wfc, bfc, out);
}